// GaussianBaseModel_2740189135627
// MI455X (gfx1250) — compile-verified
//
#include <hip/hip_runtime.h>

// N-point Gaussian 3D covariance: out[i] = upper-tri of (R(q) * diag(s)) * (R(q) * diag(s))^T
// Memory-bound streaming kernel: 52 B/point moved once, NT cache hints, wave32.

typedef float v2f __attribute__((ext_vector_type(2)));
typedef float v4f __attribute__((ext_vector_type(4)));

#define MAX_SCALING 100.0f

__global__ __launch_bounds__(256)
void gaussian_cov3d_kernel(const float* __restrict__ scaling,   // (N,3)
                           const v4f*  __restrict__ rotation4,  // (N,4) as v4f
                           float* __restrict__ out,             // (N,6)
                           int n)
{
    int i = blockIdx.x * blockDim.x + threadIdx.x;
    if (i >= n) return;

    // ---- loads (non-temporal: single-use stream, working set >> 192MB L2) ----
    v4f q = __builtin_nontemporal_load(rotation4 + i);          // global_load_b128 th:NT

    const float* sp = scaling + 3 * (size_t)i;
    float s0 = __builtin_nontemporal_load(sp + 0);
    float s1 = __builtin_nontemporal_load(sp + 1);
    float s2 = __builtin_nontemporal_load(sp + 2);

    // ---- s = clip(exp(scaling), 0, 100); exp >= 0 so only upper clip needed ----
    float sx = fminf(__expf(s0) < 0.f ? 0.f : __expf(s0), MAX_SCALING);
    // (use precise expf for all three to match reference closely)
    sx       = fminf(expf(s0), MAX_SCALING);
    float sy = fminf(expf(s1), MAX_SCALING);
    float sz = fminf(expf(s2), MAX_SCALING);

    // ---- quaternion: fold normalization into inv = 1/|q|^2 (all R terms quadratic) ----
    float r = q.x, x = q.y, y = q.z, z = q.w;
    float nrm2 = r * r + x * x + y * y + z * z;
    float two  = 2.0f / nrm2;                                   // = 2 * inv

    float R00 = 1.0f - two * (y * y + z * z);
    float R01 =        two * (x * y - r * z);
    float R02 =        two * (x * z + r * y);
    float R10 =        two * (x * y + r * z);
    float R11 = 1.0f - two * (x * x + z * z);
    float R12 =        two * (y * z - r * x);
    float R20 =        two * (x * z - r * y);
    float R21 =        two * (y * z + r * x);
    float R22 = 1.0f - two * (x * x + y * y);

    // ---- cov = (R diag(s)) (R diag(s))^T ; cov[i][k] = sum_j R[i][j] R[k][j] s_j^2 ----
    float sx2 = sx * sx, sy2 = sy * sy, sz2 = sz * sz;

    float c00 = R00 * R00 * sx2 + R01 * R01 * sy2 + R02 * R02 * sz2;
    float c01 = R00 * R10 * sx2 + R01 * R11 * sy2 + R02 * R12 * sz2;
    float c02 = R00 * R20 * sx2 + R01 * R21 * sy2 + R02 * R22 * sz2;
    float c11 = R10 * R10 * sx2 + R11 * R11 * sy2 + R12 * R12 * sz2;
    float c12 = R10 * R20 * sx2 + R11 * R21 * sy2 + R12 * R22 * sz2;
    float c22 = R20 * R20 * sx2 + R21 * R21 * sy2 + R22 * R22 * sz2;

    // ---- store 24 B/point, 8B-aligned -> three global_store_b64 th:NT ----
    v2f* o2 = (v2f*)(out + 6 * (size_t)i);
    v2f p0; p0.x = c00; p0.y = c01;
    v2f p1; p1.x = c02; p1.y = c11;
    v2f p2; p2.x = c12; p2.y = c22;
    __builtin_nontemporal_store(p0, o2 + 0);
    __builtin_nontemporal_store(p1, o2 + 1);
    __builtin_nontemporal_store(p2, o2 + 2);
}

extern "C" void kernel_launch(void* const* d_in, const int* in_sizes, int n_in,
                              void* d_out, int out_size, void* d_ws, size_t ws_size,
                              hipStream_t stream) {
    const float* scaling   = (const float*)d_in[0];   // (N,3) float32
    const v4f*   rotation4 = (const v4f*)d_in[1];     // (N,4) float32, 16B/point
    float*       out       = (float*)d_out;           // (N,6) float32

    int n = in_sizes[0] / 3;
    const int block = 256;                            // 8 wave32 per block
    int grid = (n + block - 1) / block;

    gaussian_cov3d_kernel<<<grid, block, 0, stream>>>(scaling, rotation4, out, n);
}